// TripletLoss_78288663871570
// MI455X (gfx1250) — compile-verified
//
#include <hip/hip_runtime.h>

// Triplet margin loss, reduced to scalar mean.
//   distances:    (16384, 4096) f32
//   similarities: (16384, 4096) f32
//   permutations: (16384, 3)    int   (harness passes integer inputs as int32)
//   out: single f32 = sum(max(sign(s1>=s2)*(d1-d2) + 100, 0)) / 16384
//
// Strategy (MI455X, wave32):
//  - 1 thread per row; 512 waves -> latency hiding for the 4 random gathers.
//  - Wave-level reduction via V_WMMA_F32_16X16X4_F32 against an all-ones B
//    matrix: D[m][n] = loss[m] + loss[m+16], so an 8-element per-lane VGPR sum
//    + 2 atomics per wave completes the reduction.
//  - Tiny zeroing kernel first (d_out is poisoned; atomics accumulate).

typedef __attribute__((ext_vector_type(2))) float v2f;
typedef __attribute__((ext_vector_type(8))) float v8f;

#define N_ROWS 16384
#define M_COLS 4096
#define MARGIN 100.0f

__global__ void __launch_bounds__(64) tl_zero_kernel(float* __restrict__ out) {
    if (threadIdx.x == 0) out[0] = 0.0f;
}

__global__ void __launch_bounds__(256) TripletLoss_78288663871570_kernel(
    const float* __restrict__ distances,
    const float* __restrict__ similarities,
    const int*   __restrict__ permutations,
    float* __restrict__ out)
{
    const int row = blockIdx.x * blockDim.x + threadIdx.x;

    float loss = 0.0f;
    if (row < N_ROWS) {
        // Coalesced-ish index loads (stride-3 ints across adjacent lanes).
        const int i1 = permutations[row * 3 + 1];
        const int i2 = permutations[row * 3 + 2];

        const float* __restrict__ dr = distances    + (size_t)row * M_COLS;
        const float* __restrict__ sr = similarities + (size_t)row * M_COLS;

        // 4 independent gathers -> compiler issues all global_load_b32 before
        // a single s_wait_loadcnt; pure latency, overlapped across 512 waves.
        const float d1 = dr[i1];
        const float d2 = dr[i2];
        const float s1 = sr[i1];
        const float s2 = sr[i2];

        const float diff = (s1 >= s2) ? (d1 - d2) : (d2 - d1);
        // Pre-scale by 1/N so the atomic accumulation directly yields the mean.
        loss = fmaxf(diff + MARGIN, 0.0f) * (1.0f / (float)N_ROWS);
    }

    // ---- wave32 reduction via WMMA (EXEC all-1s here: control flow has
    // reconverged, grid is an exact multiple of 32) ----
    // A (16x4 f32, 2 VGPRs/lane): VGPR0 = lane's loss (K=0 for lanes 0-15,
    // K=2 for lanes 16-31), VGPR1 = 0 (K=1 / K=3).
    v2f a; a.x = loss; a.y = 0.0f;
    // B (4x16 f32) = all ones -> D[m][n] = sum_k A[m][k] = loss[m] + loss[m+16].
    v2f b; b.x = 1.0f; b.y = 1.0f;
    v8f c = {};
    c = __builtin_amdgcn_wmma_f32_16x16x4_f32(
            /*neg_a=*/false, a, /*neg_b=*/false, b,
            /*c_mod=*/(short)0, c, /*reuse_a=*/false, /*reuse_b=*/false);

    // Per-lane sum of the 8 D rows it holds:
    //   lanes 0-15  -> sum over m=0..7  of (loss[m] + loss[m+16])
    //   lanes 16-31 -> sum over m=8..15 of (loss[m] + loss[m+16])
    float half_sum = ((c[0] + c[1]) + (c[2] + c[3]))
                   + ((c[4] + c[5]) + (c[6] + c[7]));

    const int lane = threadIdx.x & 31;
    if (lane == 0 || lane == 16) {
        atomicAdd(out, half_sum);   // global_atomic_add_f32; 2 per wave
    }
}

extern "C" void kernel_launch(void* const* d_in, const int* in_sizes, int n_in,
                              void* d_out, int out_size, void* d_ws, size_t ws_size,
                              hipStream_t stream) {
    (void)in_sizes; (void)n_in; (void)out_size; (void)d_ws; (void)ws_size;

    const float* distances    = (const float*)d_in[0];
    const float* similarities = (const float*)d_in[1];
    const int*   permutations = (const int*)  d_in[2];
    float*       out          = (float*)d_out;

    // Zero the accumulator first (same stream -> ordered; graph-capture safe).
    tl_zero_kernel<<<1, 64, 0, stream>>>(out);

    // One thread per row: 16384 threads = 64 blocks x 256 (8 wave32s/block).
    TripletLoss_78288663871570_kernel<<<N_ROWS / 256, 256, 0, stream>>>(
        distances, similarities, permutations, out);
}